// LinearRouter_26379689132708
// MI455X (gfx1250) — compile-verified
//
#include <hip/hip_runtime.h>
#include <math.h>

typedef __attribute__((ext_vector_type(2))) float v2f;
typedef __attribute__((ext_vector_type(8))) float v8f;

#define N_TOKENS 65536
#define D_MODEL  1024
#define N_EXP    64
#define TOP_K    8

#define ROWS_PER_BLOCK 256
#define THREADS        256   // 8 waves (wave32)
#define LDS_STRIDE     67    // odd stride -> conflict-free LDS access patterns

__global__ __launch_bounds__(THREADS)
void router_wmma_kernel(const float* __restrict__ h,
                        const float* __restrict__ W,
                        const unsigned char* __restrict__ tmask,
                        float* __restrict__ out)
{
    __shared__ float lds[ROWS_PER_BLOCK * LDS_STRIDE];

    const int tid  = threadIdx.x;
    const int lane = tid & 31;
    const int wave = tid >> 5;
    const int l16  = lane & 15;
    const int lhi  = lane >> 4;            // 0 for lanes 0-15, 1 for 16-31
    const int blockRow = blockIdx.x * ROWS_PER_BLOCK;

    // ---------------- Phase 1: logits = h @ W^T via V_WMMA_F32_16X16X4_F32 ----
    // Each wave: 32 rows x 64 experts = 2 row-tiles x 4 expert-tiles (16x16 each).
    const int waveRow = blockRow + wave * 32;

    // A frags: M = l16, per-lane K pair = k0 + lhi*2 + {0,1}  (ISA 16x4 f32 layout)
    const float* pA0 = h + (size_t)(waveRow + l16) * D_MODEL + (lhi << 1);
    const float* pA1 = pA0 + (size_t)16 * D_MODEL;
    // B frags: N = expert = t*16 + l16, per-lane K pair = k0 + lhi*2 + {0,1}
    const float* pB0 = W + (size_t)(l16 +  0) * D_MODEL + (lhi << 1);
    const float* pB1 = W + (size_t)(l16 + 16) * D_MODEL + (lhi << 1);
    const float* pB2 = W + (size_t)(l16 + 32) * D_MODEL + (lhi << 1);
    const float* pB3 = W + (size_t)(l16 + 48) * D_MODEL + (lhi << 1);

    v8f acc[2][4];
#pragma unroll
    for (int s = 0; s < 2; ++s)
#pragma unroll
        for (int t = 0; t < 4; ++t)
            acc[s][t] = (v8f)0.0f;

    for (int k0 = 0; k0 < D_MODEL; k0 += 4) {
        v2f a0 = *(const v2f*)(pA0 + k0);
        v2f a1 = *(const v2f*)(pA1 + k0);
        v2f b0 = *(const v2f*)(pB0 + k0);
        v2f b1 = *(const v2f*)(pB1 + k0);
        v2f b2 = *(const v2f*)(pB2 + k0);
        v2f b3 = *(const v2f*)(pB3 + k0);
        // 8 args: (neg_a, A, neg_b, B, c_mod, C, reuse_a, reuse_b)
        acc[0][0] = __builtin_amdgcn_wmma_f32_16x16x4_f32(false, a0, false, b0, (short)0, acc[0][0], false, false);
        acc[0][1] = __builtin_amdgcn_wmma_f32_16x16x4_f32(false, a0, false, b1, (short)0, acc[0][1], false, false);
        acc[0][2] = __builtin_amdgcn_wmma_f32_16x16x4_f32(false, a0, false, b2, (short)0, acc[0][2], false, false);
        acc[0][3] = __builtin_amdgcn_wmma_f32_16x16x4_f32(false, a0, false, b3, (short)0, acc[0][3], false, false);
        acc[1][0] = __builtin_amdgcn_wmma_f32_16x16x4_f32(false, a1, false, b0, (short)0, acc[1][0], false, false);
        acc[1][1] = __builtin_amdgcn_wmma_f32_16x16x4_f32(false, a1, false, b1, (short)0, acc[1][1], false, false);
        acc[1][2] = __builtin_amdgcn_wmma_f32_16x16x4_f32(false, a1, false, b2, (short)0, acc[1][2], false, false);
        acc[1][3] = __builtin_amdgcn_wmma_f32_16x16x4_f32(false, a1, false, b3, (short)0, acc[1][3], false, false);
    }

    // Spill tiles to LDS. C/D layout: VGPR j -> row (j | lhi*8), col = l16.
#pragma unroll
    for (int s = 0; s < 2; ++s)
#pragma unroll
        for (int t = 0; t < 4; ++t)
#pragma unroll
            for (int j = 0; j < 8; ++j) {
                int lrow = wave * 32 + s * 16 + (lhi << 3) + j;
                lds[lrow * LDS_STRIDE + t * 16 + l16] = acc[s][t][j];
            }

    __syncthreads();

    // ---------------- Phase 2: softmax + top-8 + renorm, one thread per token --
    const int r = blockRow + tid;
    const float* Lr = &lds[tid * LDS_STRIDE];

    float l[N_EXP];
#pragma unroll
    for (int j = 0; j < N_EXP; ++j) l[j] = Lr[j];

    const bool valid = (tmask[r] != 0);

    // Iterative argmax (strict '>' + ascending index == jax top_k tie-break).
    unsigned long long chosen = 0ull;
    float m0 = 0.0f, S8 = 0.0f;
    for (int p = 0; p < TOP_K; ++p) {
        float best = -__builtin_inff();
        int   bi   = 0;
#pragma unroll
        for (int j = 0; j < N_EXP; ++j) {
            float vj = ((chosen >> j) & 1ull) ? -__builtin_inff() : l[j];
            if (vj > best) { best = vj; bi = j; }
        }
        if (p == 0) m0 = best;
        chosen |= (1ull << bi);
        S8 += __expf(best - m0);      // probs = exp(l-m0) / sum_top8 exp(l-m0)
    }
    const float invS8 = 1.0f / S8;    // EPS_D clip can never bind for valid tokens

    const size_t TOT = (size_t)N_TOKENS * N_EXP;
    float* om = out + (size_t)r * N_EXP;          // mask_full (as 0/1 f32)
    float* op = om + TOT;                         // probs
    float* ol = op + TOT;                         // logits_clean
    float* os = ol + TOT;                         // logits_sel

#pragma unroll
    for (int j = 0; j < N_EXP; j += 4) {
        float fm[4], fp[4], fl[4], fs[4];
#pragma unroll
        for (int c = 0; c < 4; ++c) {
            int e = j + c;
            bool sel = valid && (((chosen >> e) & 1ull) != 0ull);
            float le = l[e];
            fm[c] = sel ? 1.0f : 0.0f;
            fp[c] = sel ? __expf(le - m0) * invS8 : 0.0f;
            fl[c] = le;
            fs[c] = valid ? le : -__builtin_inff();
        }
        *(float4*)(om + j) = make_float4(fm[0], fm[1], fm[2], fm[3]);
        *(float4*)(op + j) = make_float4(fp[0], fp[1], fp[2], fp[3]);
        *(float4*)(ol + j) = make_float4(fl[0], fl[1], fl[2], fl[3]);
        *(float4*)(os + j) = make_float4(fs[0], fs[1], fs[2], fs[3]);
    }
}

extern "C" void kernel_launch(void* const* d_in, const int* in_sizes, int n_in,
                              void* d_out, int out_size, void* d_ws, size_t ws_size,
                              hipStream_t stream) {
    const float* h = (const float*)d_in[0];
    const float* W = (const float*)d_in[1];
    const unsigned char* tmask = (const unsigned char*)d_in[2];  // numpy bool = 1 byte
    float* out = (float*)d_out;

    dim3 grid(N_TOKENS / ROWS_PER_BLOCK);
    dim3 block(THREADS);
    router_wmma_kernel<<<grid, block, 0, stream>>>(h, W, tmask, out);
}